// RelativePositionEncoding_25305947308291
// MI455X (gfx1250) — compile-verified
//
#include <hip/hip_runtime.h>

// ---------------------------------------------------------------------------
// RelativePositionEncoding for MI455X (gfx1250, wave32).
//
// p[i,j,:] = W_pos[d_res(i,j)] + W_tok[d_tok(i,j)] + T_chain[d_ch(i,j)]
// where T_chain folds the entity term:  T[k] = W_chain[k] + w_ent (k<5),
//                                       T[5] = W_chain[5]            (b_ent==0 case)
// Output is 1.21 GB of streaming f32 stores -> HBM-store bound (~52us floor).
// Tables (71 KB) are staged in LDS; gathers are ds_load_b128; stores are
// 512B-coalesced nontemporal global_store_b128 per wave per j.
// T_chain is built once with an exact-f32 V_WMMA_F32_16X16X4_F32.
// ---------------------------------------------------------------------------

#define CZ        128           // C_Z channels
#define NPOS      66            // 2*(R_MAX+1)
#define ROWS_W    132           // W_pos rows + W_tok rows (contiguous in W)
#define TCH_ROWS  8             // chain table padded 6 -> 8 rows

typedef float v2f __attribute__((ext_vector_type(2)));
typedef float v4f __attribute__((ext_vector_type(4)));
typedef float v8f __attribute__((ext_vector_type(8)));

// ---------------------------------------------------------------------------
// Kernel 1: fold w_ent into the 6-row chain table via exact f32 WMMA.
//   D(16x16) = A(16x4) x B(4x16) + C(16x16)
//   A[m][0] = (m<5) ? 1 : 0      (only K=0 column nonzero; rest zeroed)
//   B[0][n] = w_ent[n]           (only K=0 row nonzero; rest zeroed)
//   C[m][n] = W_chain[m][n] (m<6), 0 otherwise
// => D[m][n] = W_chain[m][n] + (m<5)*w_ent[n]   -- exact in f32 (1.0*x is exact)
// One wave, 8 channel tiles of 16.
// ---------------------------------------------------------------------------
__global__ void relpos_prep_chain_wmma(const float* __restrict__ W,
                                       float* __restrict__ Tch) {
  const int lane = threadIdx.x;                 // 0..31 (one full wave; EXEC all 1s at WMMA)
  const float* w_ent   = W + 132 * CZ;          // W[2*N_POS]
  const float* W_chain = W + 133 * CZ;          // W[2*N_POS+1 : ]
  const int col = lane & 15;

#pragma unroll
  for (int ct = 0; ct < 8; ++ct) {
    const int n = col + ct * 16;

    // C: 16x16 f32 layout -> VGPR v holds row m=v (lanes 0-15) / m=v+8 (lanes 16-31)
    v8f c = {0.f, 0.f, 0.f, 0.f, 0.f, 0.f, 0.f, 0.f};
    if (lane < 16) {
#pragma unroll
      for (int v = 0; v < 6; ++v) c[v] = W_chain[v * CZ + n];
    }

    // A: 16x4 f32 -> VGPR0 lanes 0-15 hold (M=lane, K=0). Everything else zero.
    v2f a = {0.f, 0.f};
    if (lane < 5) a.x = 1.0f;

    // B: 4x16 f32 -> VGPR0 lanes 0-15 hold (K=0, N=lane). Everything else zero.
    v2f b = {0.f, 0.f};
    if (lane < 16) b.x = w_ent[n];

    v8f d = __builtin_amdgcn_wmma_f32_16x16x4_f32(
        /*neg_a=*/false, a, /*neg_b=*/false, b,
        /*c_mod=*/(short)0, c, /*reuse_a=*/false, /*reuse_b=*/false);

    if (lane < 16) {
#pragma unroll
      for (int v = 0; v < 8; ++v) Tch[v * CZ + n] = d[v];  // rows 0..7 (6,7 stay zero)
    }
  }
}

// ---------------------------------------------------------------------------
// Kernel 2: streaming gather-sum. Block = 256 threads = 8 waves.
// Wave w handles row i = blockIdx.x*8 + w over j-range [blockIdx.y*jchunk, ...).
// Lane owns channels [4*lane, 4*lane+4) -> one v4f per (i,j).
// ---------------------------------------------------------------------------
__global__ __launch_bounds__(256) void relpos_main(
    const int* __restrict__ asym, const int* __restrict__ resi,
    const int* __restrict__ ent,  const int* __restrict__ tok,
    const int* __restrict__ sym,  const float* __restrict__ W,
    const float* __restrict__ Tch, float* __restrict__ out,
    int N, int jchunk) {
  __shared__ __align__(16) float smem[(ROWS_W + TCH_ROWS) * CZ];  // 71680 B < 320 KB/WGP

  // --- stage W_pos|W_tok (contiguous 132 rows) and folded chain table into LDS
  const int tid = threadIdx.x;
  {
    const v4f* Wg = (const v4f*)W;
    v4f* S4 = (v4f*)smem;
    for (int idx = tid; idx < ROWS_W * CZ / 4; idx += 256) S4[idx] = Wg[idx];
    const v4f* Tg = (const v4f*)Tch;
    for (int idx = tid; idx < TCH_ROWS * CZ / 4; idx += 256)
      S4[ROWS_W * CZ / 4 + idx] = Tg[idx];
  }
  __syncthreads();

  const int wave = tid >> 5;
  const int lane = tid & 31;
  const int i = blockIdx.x * 8 + wave;
  if (i >= N) return;

  const int j0 = blockIdx.y * jchunk;
  int j1 = j0 + jchunk;
  if (j1 > N) j1 = N;
  if (j0 >= j1) return;

  // row-i attributes: wave-uniform scalars
  const int ai = asym[i], ri = resi[i], ei = ent[i], ti = tok[i], si = sym[i];

  const float* s_pos = smem;
  const float* s_tok = smem + NPOS * CZ;
  const float* s_ch  = smem + ROWS_W * CZ;

  v4f* op = (v4f*)out + ((size_t)i * (size_t)N + (size_t)j0) * (CZ / 4) + lane;

#pragma unroll 2
  for (int j = j0; j < j1; ++j) {
    // uniform (SALU) index math
    const int aj = asym[j], rj = resi[j], ej = ent[j], tj = tok[j], sj = sym[j];

    int dr = ri - rj + 32;
    dr = dr < 0 ? 0 : (dr > 64 ? 64 : dr);
    if (ai != aj) dr = 65;

    int dt = ti - tj + 32;
    dt = dt < 0 ? 0 : (dt > 64 ? 64 : dt);
    if (ai != aj || ri != rj) dt = 65;

    int dc = si - sj + 2;
    dc = dc < 0 ? 0 : (dc > 4 ? 4 : dc);
    if (ei != ej) dc = 5;

    // three ds_load_b128 gathers (rows are 512B, lanes read consecutive banks)
    const v4f vp = ((const v4f*)(s_pos + dr * CZ))[lane];
    const v4f vt = ((const v4f*)(s_tok + dt * CZ))[lane];
    const v4f vc = ((const v4f*)(s_ch  + dc * CZ))[lane];

    const v4f r = vp + vt + vc;

    // 512B contiguous per wave; streaming output -> non-temporal
    __builtin_nontemporal_store(r, op);
    op += 32;
  }
}

// ---------------------------------------------------------------------------
extern "C" void kernel_launch(void* const* d_in, const int* in_sizes, int n_in,
                              void* d_out, int out_size, void* d_ws, size_t ws_size,
                              hipStream_t stream) {
  const int*   asym = (const int*)d_in[0];
  const int*   resi = (const int*)d_in[1];
  const int*   ent  = (const int*)d_in[2];
  const int*   tok  = (const int*)d_in[3];
  const int*   sym  = (const int*)d_in[4];
  const float* W    = (const float*)d_in[5];
  float* out = (float*)d_out;
  float* Tch = (float*)d_ws;                 // 8*128 f32 = 4 KB scratch
  const int N = in_sizes[0];                 // 1536

  // 1) fold entity vector into 6-row chain table (exact f32, via WMMA)
  relpos_prep_chain_wmma<<<dim3(1), dim3(32), 0, stream>>>(W, Tch);

  // 2) bandwidth-bound gather-sum
  const int JSPLIT = 6;
  const int jchunk = (N + JSPLIT - 1) / JSPLIT;
  dim3 grid((N + 7) / 8, JSPLIT);
  relpos_main<<<grid, dim3(256), 0, stream>>>(asym, resi, ent, tok, sym, W, Tch,
                                              out, N, jchunk);
}